// RoadGNN_53163105190455
// MI455X (gfx1250) — compile-verified
//
#include <hip/hip_runtime.h>
#include <hip/hip_bf16.h>
#include <stdint.h>

// ---------------------------------------------------------------------------
// RoadGNN (3-layer GraphSAGE + BN + residual) for MI455X (gfx1250, wave32).
//
// Layer plan:
//   deg/invdeg once.
//   L1: agg1[N,2] via edge atomics; z1 = (agg1*inv)@W1l + b1 + x@W1r  (tiny K=2,
//       plain VALU FMAs); BN stats fused; x1 = relu(bn(z1)).
//   L2: agg2[N,64] via edge atomics; z2 = [agg2*inv | x1] @ [W2l;W2r] + b2
//       computed with V_WMMA_F32_16X16X4_F32 (fp32 tensor path -> full
//       reference accuracy, still >> memory roofline); BN stats fused into the
//       GEMM epilogue; x2 = relu(bn(z2)) + x1.
//   L3: by linearity, project first: sl=x2@W3l, sr=x2@W3r (scalars), then
//       aggregate sl over edges (1 float/edge instead of 64).
//       out = sum3*inv + b3 + sr.
//
// edge_index cast per harness contract: integer inputs -> const int*.
// Workspace use: ~196*N floats (~392 MB) with buffer reuse.
// ---------------------------------------------------------------------------

typedef __attribute__((ext_vector_type(2))) float v2f;
typedef __attribute__((ext_vector_type(8))) float v8f;

#define LDT 132  // padded LDS row stride (dwords) for transposed weights

// ---------------- degree + layer-1 aggregation (per edge) -------------------
__global__ void edge_l1_kernel(const int* __restrict__ ei, const float* __restrict__ x,
                               float* __restrict__ deg, float* __restrict__ agg1, int E) {
    int e = blockIdx.x * blockDim.x + threadIdx.x;
    if (e >= E) return;
    int src = ei[e];
    int dst = ei[E + e];
    atomicAdd(&deg[dst], 1.0f);
    atomicAdd(&agg1[2 * dst + 0], x[2 * src + 0]);
    atomicAdd(&agg1[2 * dst + 1], x[2 * src + 1]);
}

// ---------------- layer-1 dense (K=2) + invdeg + BN stats -------------------
__global__ __launch_bounds__(256)
void l1_z_kernel(const float* __restrict__ x, const float* __restrict__ agg1,
                 const float* __restrict__ deg,
                 const float* __restrict__ W1l, const float* __restrict__ b1,
                 const float* __restrict__ W1r,
                 float* __restrict__ invdeg, float* __restrict__ z1,
                 float* __restrict__ gsum, float* __restrict__ gsumsq, int n) {
    __shared__ float red[256], redq[256];
    int tid = threadIdx.x;
    long long start = (long long)blockIdx.x * blockDim.x + tid;
    long long stride = (long long)gridDim.x * blockDim.x;  // multiple of 64
    long long total = (long long)n * 64;
    int h = (int)(start & 63);  // fixed channel per thread
    float wl0 = W1l[h], wl1 = W1l[64 + h];
    float wr0 = W1r[h], wr1 = W1r[64 + h];
    float bb  = b1[h];
    float s = 0.0f, sq = 0.0f;
    for (long long i = start; i < total; i += stride) {
        int node = (int)(i >> 6);
        float inv = 1.0f / fmaxf(deg[node], 1.0f);
        if (h == 0) invdeg[node] = inv;
        float a0 = agg1[2 * node + 0] * inv;
        float a1 = agg1[2 * node + 1] * inv;
        float x0 = x[2 * node + 0];
        float x1v = x[2 * node + 1];
        float z = fmaf(a0, wl0, fmaf(a1, wl1, fmaf(x0, wr0, fmaf(x1v, wr1, bb))));
        z1[i] = z;
        s += z; sq += z * z;
    }
    red[tid] = s; redq[tid] = sq;
    __syncthreads();
    if (tid < 64) {  // lanes tid, tid+64, ... all share channel tid
        float ts = red[tid] + red[tid + 64] + red[tid + 128] + red[tid + 192];
        float tq = redq[tid] + redq[tid + 64] + redq[tid + 128] + redq[tid + 192];
        atomicAdd(&gsum[tid], ts);
        atomicAdd(&gsumsq[tid], tq);
    }
}

// ---------------- BN finalize: mean/var -> scale/shift ----------------------
__global__ void bn_fin_kernel(const float* __restrict__ gsum, const float* __restrict__ gsumsq,
                              const float* __restrict__ g, const float* __restrict__ be,
                              float* __restrict__ scale, float* __restrict__ shift, float invn) {
    int t = threadIdx.x;
    if (t < 64) {
        float m = gsum[t] * invn;
        float v = gsumsq[t] * invn - m * m;         // biased var (matches ref)
        float r = rsqrtf(fmaxf(v, 0.0f) + 1e-5f);
        float sc = r * g[t];
        scale[t] = sc;
        shift[t] = be[t] - m * sc;
    }
}

// ---------------- apply BN + ReLU (float4) ----------------------------------
__global__ void bnrelu_kernel(const float* __restrict__ z, const float* __restrict__ scale,
                              const float* __restrict__ shift, float* __restrict__ out,
                              long long total4) {
    long long i = (long long)blockIdx.x * blockDim.x + threadIdx.x;
    if (i >= total4) return;
    int h = (int)((i * 4) & 63);
    float4 zv = ((const float4*)z)[i];
    float4 o;
    o.x = fmaxf(fmaf(zv.x, scale[h + 0], shift[h + 0]), 0.0f);
    o.y = fmaxf(fmaf(zv.y, scale[h + 1], shift[h + 1]), 0.0f);
    o.z = fmaxf(fmaf(zv.z, scale[h + 2], shift[h + 2]), 0.0f);
    o.w = fmaxf(fmaf(zv.w, scale[h + 3], shift[h + 3]), 0.0f);
    ((float4*)out)[i] = o;
}

// ---------------- apply BN + ReLU + residual --------------------------------
__global__ void bnrelu_res_kernel(const float* __restrict__ z, const float* __restrict__ scale,
                                  const float* __restrict__ shift, const float* __restrict__ resid,
                                  float* __restrict__ out, long long total4) {
    long long i = (long long)blockIdx.x * blockDim.x + threadIdx.x;
    if (i >= total4) return;
    int h = (int)((i * 4) & 63);
    float4 zv = ((const float4*)z)[i];
    float4 rv = ((const float4*)resid)[i];
    float4 o;
    o.x = fmaxf(fmaf(zv.x, scale[h + 0], shift[h + 0]), 0.0f) + rv.x;
    o.y = fmaxf(fmaf(zv.y, scale[h + 1], shift[h + 1]), 0.0f) + rv.y;
    o.z = fmaxf(fmaf(zv.z, scale[h + 2], shift[h + 2]), 0.0f) + rv.z;
    o.w = fmaxf(fmaf(zv.w, scale[h + 3], shift[h + 3]), 0.0f) + rv.w;
    ((float4*)out)[i] = o;
}

// ---------------- layer-2 edge scatter (64 channels) ------------------------
__global__ void edge_l2_kernel(const int* __restrict__ ei, const float* __restrict__ x1,
                               float* __restrict__ agg2, int E) {
    long long t = (long long)blockIdx.x * blockDim.x + threadIdx.x;
    if (t >= (long long)E * 16) return;
    int e = (int)(t >> 4);
    int q = ((int)t & 15) * 4;
    int src = ei[e];
    int dst = ei[E + e];
    float4 v = *(const float4*)(x1 + (size_t)src * 64 + q);
    float* d = agg2 + (size_t)dst * 64 + q;
    atomicAdd(d + 0, v.x);
    atomicAdd(d + 1, v.y);
    atomicAdd(d + 2, v.z);
    atomicAdd(d + 3, v.w);
}

// ---------------- layer-2 GEMM via V_WMMA_F32_16X16X4_F32 -------------------
// z2[N,64] = [agg2*invdeg | x1] (N x 128) @ [W2l ; W2r] (128 x 64) + b2
// 8 waves/block, each wave owns a 16x64 tile = 4 f32 16x16 accumulators.
// A layout (32-bit A 16x4): lanes 0-15 hold K={k,k+1}, lanes 16-31 K={k+2,k+3}.
// B layout mirrors A: v0 = B[k0][col], v1 = B[k0+1][col], col = 16j + (lane&15).
// BN column stats fused into epilogue (LDS partials -> 64 global atomics/blk).
__global__ __launch_bounds__(256)
void gemm_l2_wmma_kernel(const float* __restrict__ agg2, const float* __restrict__ x1,
                         const float* __restrict__ invdeg,
                         const float* __restrict__ W2l, const float* __restrict__ W2r,
                         const float* __restrict__ bias2, float* __restrict__ z2,
                         float* __restrict__ gsum, float* __restrict__ gsumsq, int n) {
    __shared__ float WcT[64 * LDT];   // transposed combined weights: WcT[c][k]
    __shared__ float s_sum[64];
    __shared__ float s_sq[64];

    int tid = threadIdx.x;
    for (int i = tid; i < 64 * 128; i += 256) {
        int k = i >> 6;   // 0..127
        int c = i & 63;   // 0..63
        float w = (k < 64) ? W2l[k * 64 + c] : W2r[(k - 64) * 64 + c];
        WcT[c * LDT + k] = w;
    }
    if (tid < 64) { s_sum[tid] = 0.0f; s_sq[tid] = 0.0f; }
    __syncthreads();

    int wave = tid >> 5;
    int lane = tid & 31;
    int half = lane >> 4;   // 0 or 1
    int m    = lane & 15;
    int r0   = blockIdx.x * 128 + wave * 16;
    int rowA = r0 + m;
    int rowAc = rowA < n ? rowA : (n - 1);         // clamp (tail garbage masked later)
    const float* aggRow = agg2 + (size_t)rowAc * 64;
    const float* xRow   = x1   + (size_t)rowAc * 64;
    float inv = invdeg[rowAc];

    v8f acc[4] = {};

    for (int k = 0; k < 128; k += 4) {
        int k0 = k + 2 * half;  // K boundary (64) is a multiple of 4 -> branch is uniform
        v2f a;
        if (k < 64) {
            a.x = aggRow[k0] * inv;
            a.y = aggRow[k0 + 1] * inv;
        } else {
            a.x = xRow[k0 - 64];
            a.y = xRow[k0 - 63];
        }
        const float* wt = &WcT[m * LDT + k0];
#pragma unroll
        for (int j = 0; j < 4; ++j) {
            v2f b;
            b.x = wt[j * 16 * LDT + 0];
            b.y = wt[j * 16 * LDT + 1];
            acc[j] = __builtin_amdgcn_wmma_f32_16x16x4_f32(
                false, a, false, b, (short)0, acc[j], false, false);
        }
    }

    // epilogue: bias + store + fused BN stats
#pragma unroll
    for (int j = 0; j < 4; ++j) {
        int col = 16 * j + m;
        float bj = bias2[col];
        float s = 0.0f, sq = 0.0f;
#pragma unroll
        for (int r = 0; r < 8; ++r) {
            int row = r0 + r + 8 * half;  // C/D layout: lanes 16-31 hold M = r+8
            float v = acc[j][r] + bj;
            if (row < n) {
                z2[(size_t)row * 64 + col] = v;
                s += v; sq += v * v;
            }
        }
        atomicAdd(&s_sum[col], s);
        atomicAdd(&s_sq[col], sq);
    }
    __syncthreads();
    if (tid < 64) {
        atomicAdd(&gsum[tid], s_sum[tid]);
        atomicAdd(&gsumsq[tid], s_sq[tid]);
    }
}

// ---------------- layer-3 scalar projections --------------------------------
__global__ __launch_bounds__(256)
void dot3_kernel(const float* __restrict__ x2, const float* __restrict__ W3l,
                 const float* __restrict__ W3r, float* __restrict__ sl,
                 float* __restrict__ sr, int n) {
    __shared__ float wl[64], wr[64];
    int tid = threadIdx.x;
    if (tid < 64) { wl[tid] = W3l[tid]; wr[tid] = W3r[tid]; }
    __syncthreads();
    int node = blockIdx.x * blockDim.x + tid;
    if (node >= n) return;
    const float4* row = (const float4*)(x2 + (size_t)node * 64);
    float a = 0.0f, b = 0.0f;
#pragma unroll
    for (int q = 0; q < 16; ++q) {
        float4 v = row[q];
        float4 l = ((const float4*)wl)[q];
        float4 r = ((const float4*)wr)[q];
        a += v.x * l.x + v.y * l.y + v.z * l.z + v.w * l.w;
        b += v.x * r.x + v.y * r.y + v.z * r.z + v.w * r.w;
    }
    sl[node] = a;
    sr[node] = b;
}

// ---------------- layer-3 scalar edge aggregation ---------------------------
__global__ void edge_l3_kernel(const int* __restrict__ ei, const float* __restrict__ sl,
                               float* __restrict__ sum3, int E) {
    int e = blockIdx.x * blockDim.x + threadIdx.x;
    if (e >= E) return;
    atomicAdd(&sum3[ei[E + e]], sl[ei[e]]);
}

// ---------------- final combine ---------------------------------------------
__global__ void final_kernel(const float* __restrict__ sum3, const float* __restrict__ sr,
                             const float* __restrict__ invdeg, const float* __restrict__ b3,
                             float* __restrict__ out, int n) {
    int i = blockIdx.x * blockDim.x + threadIdx.x;
    if (i >= n) return;
    out[i] = fmaf(sum3[i], invdeg[i], b3[0] + sr[i]);
}

// ---------------------------------------------------------------------------
extern "C" void kernel_launch(void* const* d_in, const int* in_sizes, int n_in,
                              void* d_out, int out_size, void* d_ws, size_t ws_size,
                              hipStream_t stream) {
    const float* x   = (const float*)d_in[0];
    const int*   ei  = (const int*)d_in[1];   // [2,E]; harness delivers ints as int32
    const float* W1l = (const float*)d_in[2];
    const float* b1  = (const float*)d_in[3];
    const float* W1r = (const float*)d_in[4];
    const float* g1  = (const float*)d_in[5];
    const float* be1 = (const float*)d_in[6];
    const float* W2l = (const float*)d_in[7];
    const float* b2  = (const float*)d_in[8];
    const float* W2r = (const float*)d_in[9];
    const float* g2  = (const float*)d_in[10];
    const float* be2 = (const float*)d_in[11];
    const float* W3l = (const float*)d_in[12];
    const float* b3  = (const float*)d_in[13];
    const float* W3r = (const float*)d_in[14];

    const int n = in_sizes[0] / 2;   // CIN = 2
    const int E = in_sizes[1] / 2;

    // ---- workspace layout (floats); total ~196*n + 512 (~392 MB) ----
    float* ws = (float*)d_ws;
    size_t o = 0;
    float* invdeg = ws + o; o += (size_t)n;
    float* deg    = ws + o; o += (size_t)n;
    float* agg1   = ws + o; o += 2 * (size_t)n;
    float* stats  = ws + o; o += 512;
    float* sum1 = stats,       *sumsq1 = stats + 64,  *scale1 = stats + 128, *shift1 = stats + 192;
    float* sum2 = stats + 256, *sumsq2 = stats + 320, *scale2 = stats + 384, *shift2 = stats + 448;
    size_t big = (size_t)n * 64;
    float* big1 = ws + o; o += big;  // z1 -> agg2 -> x2
    float* big2 = ws + o; o += big;  // x1 -> (sl, sr, sum3)
    float* big3 = ws + o; o += big;  // z2

    const int B = 256;

    // zero: deg, agg1, stats (contiguous)
    hipMemsetAsync(deg, 0, sizeof(float) * (3 * (size_t)n + 512), stream);

    // L1
    edge_l1_kernel<<<(E + B - 1) / B, B, 0, stream>>>(ei, x, deg, agg1, E);
    l1_z_kernel<<<2048, B, 0, stream>>>(x, agg1, deg, W1l, b1, W1r,
                                        invdeg, big1, sum1, sumsq1, n);
    bn_fin_kernel<<<1, 64, 0, stream>>>(sum1, sumsq1, g1, be1, scale1, shift1, 1.0f / (float)n);
    long long tot4 = (long long)n * 16;
    bnrelu_kernel<<<(int)((tot4 + B - 1) / B), B, 0, stream>>>(big1, scale1, shift1, big2, tot4);

    // L2
    hipMemsetAsync(big1, 0, big * sizeof(float), stream);   // agg2 = big1
    long long t2 = (long long)E * 16;
    edge_l2_kernel<<<(int)((t2 + B - 1) / B), B, 0, stream>>>(ei, big2, big1, E);
    gemm_l2_wmma_kernel<<<(n + 127) / 128, 256, 0, stream>>>(big1, big2, invdeg,
                                                             W2l, W2r, b2, big3,
                                                             sum2, sumsq2, n);
    bn_fin_kernel<<<1, 64, 0, stream>>>(sum2, sumsq2, g2, be2, scale2, shift2, 1.0f / (float)n);
    // x2 = relu(bn(z2)) + x1 -> big1 (agg2 dead after GEMM)
    bnrelu_res_kernel<<<(int)((tot4 + B - 1) / B), B, 0, stream>>>(big3, scale2, shift2,
                                                                   big2, big1, tot4);

    // L3 (project first, then aggregate scalars)
    float* sl   = big2;                 // x1 dead after residual apply
    float* sr   = big2 + (size_t)n;
    float* sum3 = big2 + 2 * (size_t)n;
    dot3_kernel<<<(n + B - 1) / B, B, 0, stream>>>(big1, W3l, W3r, sl, sr, n);
    hipMemsetAsync(sum3, 0, sizeof(float) * (size_t)n, stream);
    edge_l3_kernel<<<(E + B - 1) / B, B, 0, stream>>>(ei, sl, sum3, E);
    final_kernel<<<(n + B - 1) / B, B, 0, stream>>>(sum3, sr, invdeg, b3, (float*)d_out, n);
}